// SphericalBasisLayer_63367947485957
// MI455X (gfx1250) — compile-verified
//
#include <hip/hip_runtime.h>
#include <math.h>

// ---------------- constants ----------------
#define N_SPH 7
#define N_RAD 6
#define NBASIS 42            // N_SPH * N_RAD
#define CUTOFF 5.0f
// envelope p=5:  a=-(6*7)/2=-21, b=5*7=35, c=-(5*6)/2=-15
#define ENV_A (-21.0f)
#define ENV_B (35.0f)
#define ENV_C (-15.0f)

#define TPB 256
#define WS_FREQ 0
#define WS_NORM 42
#define WS_BC   84           // 8x16 B-matrix (power-major, l-minor)
#define WS_RBF  256          // rbf starts here (aligned)

typedef float v2f __attribute__((ext_vector_type(2)));
typedef float v8f __attribute__((ext_vector_type(8)));

// Legendre monomial coefficients: leg[l][k] = coef of x^k in P_l(x)
__device__ __constant__ double d_leg[7][7] = {
    { 1.0,     0.0,    0.0,     0.0,   0.0,      0.0,   0.0     },
    { 0.0,     1.0,    0.0,     0.0,   0.0,      0.0,   0.0     },
    { -0.5,    0.0,    1.5,     0.0,   0.0,      0.0,   0.0     },
    { 0.0,    -1.5,    0.0,     2.5,   0.0,      0.0,   0.0     },
    { 0.375,   0.0,   -3.75,    0.0,   4.375,    0.0,   0.0     },
    { 0.0,     1.875,  0.0,    -8.75,  0.0,      7.875, 0.0     },
    { -0.3125, 0.0,    6.5625,  0.0,  -19.6875,  0.0,  14.4375  }
};

// spherical bessel j_l(t), upward recurrence (double, matches reference)
__device__ double sjn_d(double t, int l) {
    double s = sin(t), c = cos(t);
    double j0 = s / t;
    if (l == 0) return j0;
    double j1 = s / (t * t) - c / t;
    for (int i = 2; i <= l; ++i) {
        double tmp = (2.0 * i - 1.0) / t * j1 - j0;
        j0 = j1; j1 = tmp;
    }
    return j1;
}

// ---------------- init: bessel zeros + norms + B matrix ----------------
__global__ void init_kernel(float* ws) {
    __shared__ double z[N_SPH][12];            // nz = 6 + 7 - 1 = 12
    const int tid = threadIdx.x;
    if (tid < 12) z[0][tid] = M_PI * (double)(tid + 1);
    __syncthreads();
    for (int l = 1; l < N_SPH; ++l) {
        if (tid < 12 - l) {                    // roots of this level, in parallel
            double a = z[l - 1][tid], b = z[l - 1][tid + 1];
            double fa = sjn_d(a, l);
            for (int it = 0; it < 80; ++it) {
                double m = 0.5 * (a + b);
                double fm = sjn_d(m, l);
                if (fa * fm <= 0.0) { b = m; } else { a = m; fa = fm; }
            }
            z[l][tid] = 0.5 * (a + b);
        }
        __syncthreads();
    }
    if (tid < NBASIS) {
        int l = tid / N_RAD, j = tid % N_RAD;
        double zz = z[l][j];
        ws[WS_FREQ + tid] = (float)zz;
        ws[WS_NORM + tid] = (float)(sqrt(2.0) / fabs(sjn_d(zz, l + 1)));
    }
    if (tid < 128) {                           // B[k][n] = Y_n * leg[n][k], 8x16
        int k = tid >> 4, n = tid & 15;
        double v = 0.0;
        if (n < N_SPH && k < N_SPH)
            v = sqrt((2.0 * n + 1.0) / (4.0 * M_PI)) * d_leg[n][k];
        ws[WS_BC + tid] = (float)v;
    }
}

// ---------------- kernel A: per-edge radial basis into ws ----------------
__global__ __launch_bounds__(TPB) void rbf_kernel(const float* __restrict__ dist,
                                                  const float* __restrict__ freq,
                                                  const float* __restrict__ nrm,
                                                  float* __restrict__ rbf, int E) {
    int e = blockIdx.x * TPB + threadIdx.x;
    if (e >= E) return;
    float x  = dist[e] * (1.0f / CUTOFF);            // x in [0.1, 1]
    float x2 = x * x, x4 = x2 * x2, x5 = x4 * x;
    float env = 1.0f / x + ENV_A * x5 + ENV_B * x5 * x + ENV_C * x5 * x2;

    float r[NBASIS];
#pragma unroll
    for (int l = 0; l < N_SPH; ++l) {
#pragma unroll
        for (int j = 0; j < N_RAD; ++j) {
            float t = x * freq[l * N_RAD + j];
            float it = 1.0f / t;
            float s, c;
            __sincosf(t, &s, &c);                    // args < ~35, fine
            float j0 = s * it;
            float val;
            if (l == 0) {
                val = j0;
            } else {
                float j1 = s * it * it - c * it;
#pragma unroll
                for (int i = 2; i <= l; ++i) {
                    float tmp = (2.0f * i - 1.0f) * it * j1 - j0;
                    j0 = j1; j1 = tmp;
                }
                val = j1;
            }
            r[l * N_RAD + j] = env * (nrm[l * N_RAD + j] * val);
        }
    }
    float2* o2 = (float2*)(rbf + (size_t)e * NBASIS);  // 8B aligned (42*4=168)
#pragma unroll
    for (int i = 0; i < NBASIS / 2; ++i) o2[i] = make_float2(r[2 * i], r[2 * i + 1]);
}

// ---------------- kernel B: gather + WMMA cbf + coalesced multiply-store ----
__global__ __launch_bounds__(TPB) void triplet_kernel(const float* __restrict__ angle,
                                                      const int* __restrict__ idx_kj,
                                                      const float* __restrict__ rbf,
                                                      const float* __restrict__ Bc,
                                                      float* __restrict__ out, int T) {
    __shared__ float rbf_s[TPB * NBASIS];   // 42 KB, gathered rows
    __shared__ float cth_s[TPB];
    __shared__ float cbf_s[TPB * 8];        // [t_local][l], padded to 8

    const int tid = threadIdx.x;
    const int t0  = blockIdx.x * TPB;
    const int t   = t0 + tid;
    const bool valid = (t < T);

    // ---- phase 1: cos(angle) + gather rbf row into LDS (row stays in L2) ----
    float ang = valid ? angle[t] : 0.0f;
    cth_s[tid] = __cosf(ang);
    int e = valid ? idx_kj[t] : 0;
    const float2* src2 = (const float2*)(rbf + (size_t)e * NBASIS);
    float2* dst2 = (float2*)(rbf_s + tid * NBASIS);
#pragma unroll
    for (int i = 0; i < NBASIS / 2; ++i) dst2[i] = src2[i];
    __syncthreads();

    // ---- phase 2: cbf[t,l] = sum_k cth^k * B[k][l] via V_WMMA_F32_16X16X4 ----
    const int wv   = tid >> 5;
    const int lane = tid & 31;
    const int half = lane >> 4;     // K-half / M-half selector per ISA layout
    const int ln   = lane & 15;

    // per-lane B-matrix entries (constant table in L2)
    float b0x = Bc[(half ? 2 : 0) * 16 + ln];
    float b0y = Bc[(half ? 3 : 1) * 16 + ln];
    float b1x = Bc[(half ? 6 : 4) * 16 + ln];
    float b1y = half ? 0.0f : Bc[5 * 16 + ln];   // K=7 row is zero

#pragma unroll
    for (int s = 0; s < 2; ++s) {
        int m0 = wv * 32 + s * 16;
        float c  = cth_s[m0 + ln];               // A row M=ln, both halves
        float c2 = c * c, c3 = c2 * c, c4 = c2 * c2, c5 = c4 * c, c6 = c4 * c2;
        // A 16x4 layout: lanes0-15 hold K=0,1 ; lanes16-31 hold K=2,3 (VGPR0,1)
        v2f A0 = { half ? c2 : 1.0f, half ? c3 : c   };
        v2f A1 = { half ? c6 : c4,   half ? 0.f : c5 };
        v2f B0 = { b0x, b0y };
        v2f B1 = { b1x, b1y };
        v8f acc = {0.f, 0.f, 0.f, 0.f, 0.f, 0.f, 0.f, 0.f};
        acc = __builtin_amdgcn_wmma_f32_16x16x4_f32(false, A0, false, B0,
                                                    (short)0, acc, false, false);
        acc = __builtin_amdgcn_wmma_f32_16x16x4_f32(false, A1, false, B1,
                                                    (short)0, acc, false, false);
        // D layout: VGPR v, lanes0-15 -> M=v ; lanes16-31 -> M=v+8 ; N=ln
        if (ln < N_SPH) {
            int trip = m0 + half * 8;
#pragma unroll
            for (int v = 0; v < 8; ++v)
                cbf_s[(trip + v) * 8 + ln] = acc[v];
        }
    }
    __syncthreads();

    // ---- phase 3: out[t, c] = rbf_row[c] * cbf[t, c/6], coalesced ----
    const size_t base = (size_t)t0 * NBASIS;
    const int nvalid = min(TPB, T - t0);
    if (nvalid == TPB) {
        float4* o4 = (float4*)(out + base);      // t0*42*4 % 16 == 0
        for (int i = tid; i < (TPB * NBASIS) / 4; i += TPB) {
            int m = i << 2;
            float4 r;
            float* rp = &r.x;
#pragma unroll
            for (int q = 0; q < 4; ++q) {
                int idx = m + q;
                int tl = (unsigned)idx / NBASIS;
                int cc = idx - tl * NBASIS;
                int l  = (unsigned)cc / N_RAD;
                rp[q] = rbf_s[idx] * cbf_s[(tl << 3) + l];
            }
            o4[i] = r;
        }
    } else {
        int limit = nvalid * NBASIS;
        for (int i = tid; i < limit; i += TPB) {
            int tl = (unsigned)i / NBASIS;
            int cc = i - tl * NBASIS;
            int l  = (unsigned)cc / N_RAD;
            out[base + i] = rbf_s[i] * cbf_s[(tl << 3) + l];
        }
    }
}

// ---------------- launch ----------------
extern "C" void kernel_launch(void* const* d_in, const int* in_sizes, int n_in,
                              void* d_out, int out_size, void* d_ws, size_t ws_size,
                              hipStream_t stream) {
    const float* dist   = (const float*)d_in[0];
    const float* angle  = (const float*)d_in[1];
    const int*   idx_kj = (const int*)d_in[2];
    float* out = (float*)d_out;
    float* ws  = (float*)d_ws;
    const int E = in_sizes[0];
    const int T = in_sizes[1];

    float* freq = ws + WS_FREQ;
    float* nrm  = ws + WS_NORM;
    float* Bc   = ws + WS_BC;
    float* rbf  = ws + WS_RBF;   // E*42 floats = 84 MB (fits in 192 MB L2)

    init_kernel<<<1, 128, 0, stream>>>(ws);
    rbf_kernel<<<(E + TPB - 1) / TPB, TPB, 0, stream>>>(dist, freq, nrm, rbf, E);
    triplet_kernel<<<(T + TPB - 1) / TPB, TPB, 0, stream>>>(angle, idx_kj, rbf, Bc, out, T);
}